// GNN_25958782337753
// MI455X (gfx1250) — compile-verified
//
#include <hip/hip_runtime.h>
#include <hip/hip_bf16.h>

#define GN      100000
#define GE      1600000
#define IN_F    128
#define HEADS   2
#define C1      64
#define F1      128          // HEADS*C1
#define C2      40
#define F2      80           // HEADS*C2
#define NCLS    40
#define KDIM    128          // shared K for both GEMMs
#define NEG_SLOPE 0.2f
#define BN_EPS  1e-5f

typedef float v2f __attribute__((ext_vector_type(2)));
typedef float v8f __attribute__((ext_vector_type(8)));

// ---------------- monotone float <-> uint encoding for atomic max ----------
__device__ __forceinline__ unsigned fflip(float f) {
    unsigned u = __float_as_uint(f);
    return u ^ ((unsigned)((int)u >> 31) | 0x80000000u);
}
__device__ __forceinline__ float funflip(unsigned u) {
    return __uint_as_float(u ^ (((u >> 31) - 1u) | 0x80000000u));
}
__device__ __forceinline__ float lrelu(float x) {
    return x > 0.0f ? x : x * NEG_SLOPE;
}

__device__ __forceinline__ void wait_asynccnt0() {
#if __has_builtin(__builtin_amdgcn_s_wait_asynccnt)
    __builtin_amdgcn_s_wait_asynccnt(0);
#else
    asm volatile("s_wait_asynccnt 0" ::: "memory");
#endif
}

// ---------------------------------------------------------------------------
// WMMA GEMM: C[M,Ncol] = A[M,KDIM] @ B[KDIM,Ncol], M%16==0.
// One block = one 16-row strip; 8 waves; wave w owns column tiles w, w+8,...
//  - A strip (contiguous 8KB) staged via async global->LDS b128 (ASYNCcnt).
//  - B fragments (64 floats/lane) preloaded into registers once per tile.
//  - Inner loop: 32x (ds_load_2addr + v_wmma_f32_16x16x4_f32), fp32-exact.
// ---------------------------------------------------------------------------
#if __has_builtin(__builtin_amdgcn_wmma_f32_16x16x4_f32)
#define HAVE_WMMA_F32 1
#endif

__global__ __launch_bounds__(256)
void wmma_gemm_kernel(const float* __restrict__ A, const float* __restrict__ B,
                      float* __restrict__ C, int Ncol, int nColTiles) {
    __shared__ float lA[16 * KDIM];
    const int tid = threadIdx.x;
    const int m0  = blockIdx.x * 16;

    // ---- async stage of the contiguous 16xKDIM A strip into LDS ----
    {
        const float*  gsrc  = A + (size_t)m0 * KDIM;     // strip is contiguous
        const unsigned lbase = (unsigned)(uintptr_t)lA;  // LDS-relative address
        // 16*KDIM floats = 512 float4s; 2 per thread
        for (int i = tid; i < (16 * KDIM) / 4; i += 256) {
            const unsigned lofs = lbase + (unsigned)i * 16u;
            const float*   g    = gsrc + (size_t)i * 4;
            asm volatile("global_load_async_to_lds_b128 %0, %1, off"
                         :: "v"(lofs), "v"(g) : "memory");
        }
        wait_asynccnt0();
        __syncthreads();
    }

    const int wave  = tid >> 5;
    const int lane  = tid & 31;
    const int row   = lane & 15;       // M index within tile (A) / N index (B,C)
    const int khalf = (lane >> 4) * 2; // lanes 0-15: K 0,1 ; lanes 16-31: K 2,3

    for (int ct = wave; ct < nColTiles; ct += 8) {
        const int n0  = ct * 16;
        const int col = n0 + row;

        // ---- preload this tile's B fragments into registers (L2-resident) ----
        float bx[KDIM / 4], by[KDIM / 4];
        #pragma unroll
        for (int t = 0; t < KDIM / 4; ++t) {
            bx[t] = B[(size_t)(4 * t + khalf) * Ncol + col];
            by[t] = B[(size_t)(4 * t + khalf + 1) * Ncol + col];
        }

        v8f acc = {};
        #pragma unroll
        for (int t = 0; t < KDIM / 4; ++t) {
#if HAVE_WMMA_F32
            v2f a, b;
            a.x = lA[row * KDIM + 4 * t + khalf];
            a.y = lA[row * KDIM + 4 * t + khalf + 1];
            b.x = bx[t];
            b.y = by[t];
            acc = __builtin_amdgcn_wmma_f32_16x16x4_f32(
                false, a, false, b, (short)0, acc, false, false);
#else
            for (int q = 0; q < 4; ++q)
                for (int r = 0; r < 8; ++r)
                    acc[r] += lA[(r + (lane >> 4) * 8) * KDIM + 4 * t + q] *
                              B[(size_t)(4 * t + q) * Ncol + col];
#endif
        }
        // C layout: VGPR r -> row m0 + r + (lane>=16 ? 8 : 0), col n0 + (lane&15)
        const int rbase = m0 + (lane >> 4) * 8;
        #pragma unroll
        for (int r = 0; r < 8; ++r) {
            C[(size_t)(rbase + r) * Ncol + col] = acc[r];
        }
    }
}

// ---------------------------------------------------------------------------
// Attention dot products: a_src[n,h] = sum_c h[n, h*Cp+c]*att_src[h*Cp+c]
// One wave per node; lanes 0-15 -> head 0, lanes 16-31 -> head 1.
// ---------------------------------------------------------------------------
__global__ __launch_bounds__(256)
void attn_dots_kernel(const float* __restrict__ h,
                      const float* __restrict__ att_src,
                      const float* __restrict__ att_dst,
                      float* __restrict__ a_src, float* __restrict__ a_dst,
                      int n, int F, int Cp) {
    const int wid = (blockIdx.x * blockDim.x + threadIdx.x) >> 5;
    if (wid >= n) return;
    const int lane = threadIdx.x & 31;
    const int head = lane >> 4;
    const int idx  = lane & 15;
    float ss = 0.0f, sd = 0.0f;
    for (int c = idx; c < Cp; c += 16) {
        const float hv = h[(size_t)wid * F + head * Cp + c];
        ss += hv * att_src[head * Cp + c];
        sd += hv * att_dst[head * Cp + c];
    }
    // reduce within each 16-lane half (xor masks < 16 stay inside the half)
    #pragma unroll
    for (int m = 1; m < 16; m <<= 1) {
        ss += __shfl_xor(ss, m, 32);
        sd += __shfl_xor(sd, m, 32);
    }
    if (idx == 0) {
        a_src[wid * 2 + head] = ss;
        a_dst[wid * 2 + head] = sd;
    }
}

// ---------------------------------------------------------------------------
// Edge passes. Edge id e: e < E -> (src=ei[e], dst=ei[E+e]); else self-loop.
// ---------------------------------------------------------------------------
__global__ __launch_bounds__(256)
void edge_amax_kernel(const int* __restrict__ ei,
                      const float* __restrict__ a_src,
                      const float* __restrict__ a_dst,
                      unsigned* __restrict__ amax, int E_, int n) {
    const int e = blockIdx.x * blockDim.x + threadIdx.x;
    if (e >= E_ + n) return;
    int s, d;
    if (e < E_) { s = ei[e]; d = ei[E_ + e]; } else { s = d = e - E_; }
    #pragma unroll
    for (int h = 0; h < 2; ++h) {
        const float al = lrelu(a_src[s * 2 + h] + a_dst[d * 2 + h]);
        atomicMax(&amax[d * 2 + h], fflip(al));
    }
}

__global__ __launch_bounds__(256)
void edge_denom_kernel(const int* __restrict__ ei,
                       const float* __restrict__ a_src,
                       const float* __restrict__ a_dst,
                       const unsigned* __restrict__ amax,
                       float* __restrict__ denom, int E_, int n) {
    const int e = blockIdx.x * blockDim.x + threadIdx.x;
    if (e >= E_ + n) return;
    int s, d;
    if (e < E_) { s = ei[e]; d = ei[E_ + e]; } else { s = d = e - E_; }
    #pragma unroll
    for (int h = 0; h < 2; ++h) {
        const float al = lrelu(a_src[s * 2 + h] + a_dst[d * 2 + h]);
        const float m  = funflip(amax[d * 2 + h]);
        atomicAdd(&denom[d * 2 + h], __expf(al - m));
    }
}

// One wave per edge: gather h[src] row (L2-resident), scale by softmax weight,
// atomic-add into out[dst] row.
__global__ __launch_bounds__(256)
void edge_aggregate_kernel(const int* __restrict__ ei,
                           const float* __restrict__ a_src,
                           const float* __restrict__ a_dst,
                           const unsigned* __restrict__ amax,
                           const float* __restrict__ denom,
                           const float* __restrict__ hsrc,
                           float* __restrict__ out,
                           int E_, int n, int F, int Cp) {
    const int wid = (blockIdx.x * blockDim.x + threadIdx.x) >> 5;
    if (wid >= E_ + n) return;
    const int lane = threadIdx.x & 31;
    int s, d;
    if (wid < E_) { s = ei[wid]; d = ei[E_ + wid]; } else { s = d = wid - E_; }

    const float al0 = lrelu(a_src[s * 2 + 0] + a_dst[d * 2 + 0]);
    const float al1 = lrelu(a_src[s * 2 + 1] + a_dst[d * 2 + 1]);
    const float w0  = __expf(al0 - funflip(amax[d * 2 + 0])) / denom[d * 2 + 0];
    const float w1  = __expf(al1 - funflip(amax[d * 2 + 1])) / denom[d * 2 + 1];

    const size_t sb = (size_t)s * F, db = (size_t)d * F;
    for (int f = lane; f < F; f += 32) {
        const float wv = (f < Cp) ? w0 : w1;
        atomicAdd(&out[db + f], hsrc[sb + f] * wv);
    }
}

// ---------------------------------------------------------------------------
// BatchNorm: stats then in-place apply. (BN(x + b1) == BN(x); b1 cancels.)
// ---------------------------------------------------------------------------
__global__ __launch_bounds__(128)
void bn_stats_kernel(const float* __restrict__ x, float* __restrict__ sum,
                     float* __restrict__ sumsq, int n) {
    const int f = threadIdx.x;                // 128 features
    float s = 0.0f, s2 = 0.0f;
    for (int r = blockIdx.x; r < n; r += gridDim.x) {
        const float v = x[(size_t)r * 128 + f];
        s += v; s2 += v * v;
    }
    atomicAdd(&sum[f], s);
    atomicAdd(&sumsq[f], s2);
}

__global__ __launch_bounds__(256)
void bn_apply_kernel(float* __restrict__ x, const float* __restrict__ sum,
                     const float* __restrict__ sumsq,
                     const float* __restrict__ gamma,
                     const float* __restrict__ beta, int n) {
    const size_t i = (size_t)blockIdx.x * blockDim.x + threadIdx.x;
    if (i >= (size_t)n * 128) return;
    const int f = (int)(i & 127);
    const float inv_n = 1.0f / (float)n;
    const float mean  = sum[f] * inv_n;
    const float var   = sumsq[f] * inv_n - mean * mean;
    x[i] = (x[i] - mean) * rsqrtf(var + BN_EPS) * gamma[f] + beta[f];
}

// ---------------------------------------------------------------------------
// Finalize: head-mean + b2, then log_softmax over 40 classes. Wave per node.
// ---------------------------------------------------------------------------
__global__ __launch_bounds__(256)
void finalize_kernel(const float* __restrict__ out2, const float* __restrict__ b2,
                     float* __restrict__ out, int n) {
    const int wid = (blockIdx.x * blockDim.x + threadIdx.x) >> 5;
    if (wid >= n) return;
    const int lane = threadIdx.x & 31;
    const size_t base = (size_t)wid * F2;

    const int c0 = lane, c1 = lane + 32;
    float v0 = -3.0e38f, v1 = -3.0e38f;
    if (c0 < NCLS) v0 = 0.5f * (out2[base + c0] + out2[base + NCLS + c0]) + b2[c0];
    if (c1 < NCLS) v1 = 0.5f * (out2[base + c1] + out2[base + NCLS + c1]) + b2[c1];

    float m = fmaxf(v0, v1);
    #pragma unroll
    for (int off = 1; off < 32; off <<= 1) m = fmaxf(m, __shfl_xor(m, off, 32));
    float s = 0.0f;
    if (c0 < NCLS) s += __expf(v0 - m);
    if (c1 < NCLS) s += __expf(v1 - m);
    #pragma unroll
    for (int off = 1; off < 32; off <<= 1) s += __shfl_xor(s, off, 32);
    const float lse = m + __logf(s);
    if (c0 < NCLS) out[(size_t)wid * NCLS + c0] = v0 - lse;
    if (c1 < NCLS) out[(size_t)wid * NCLS + c1] = v1 - lse;
}

// ---------------------------------------------------------------------------
extern "C" void kernel_launch(void* const* d_in, const int* in_sizes, int n_in,
                              void* d_out, int out_size, void* d_ws, size_t ws_size,
                              hipStream_t stream) {
    const float* x        = (const float*)d_in[0];
    const int*   ei       = (const int*)  d_in[1];
    const float* W1       = (const float*)d_in[2];
    const float* att_src1 = (const float*)d_in[3];
    const float* att_dst1 = (const float*)d_in[4];
    // d_in[5] = b1 (cancels under BatchNorm)
    const float* gamma    = (const float*)d_in[6];
    const float* beta     = (const float*)d_in[7];
    const float* W2       = (const float*)d_in[8];
    const float* att_src2 = (const float*)d_in[9];
    const float* att_dst2 = (const float*)d_in[10];
    const float* b2       = (const float*)d_in[11];
    float* out = (float*)d_out;

    float* ws = (float*)d_ws;
    size_t o = 0;
    float*    h1    = ws + o; o += (size_t)GN * F1;
    float*    out1  = ws + o; o += (size_t)GN * F1;
    float*    h2    = ws + o; o += (size_t)GN * F2;
    float*    out2  = ws + o; o += (size_t)GN * F2;
    float*    a_src = ws + o; o += (size_t)GN * 2;
    float*    a_dst = ws + o; o += (size_t)GN * 2;
    unsigned* amax  = (unsigned*)(ws + o); o += (size_t)GN * 2;
    float*    denom = ws + o; o += (size_t)GN * 2;
    float*    bnsum = ws + o; o += 128;
    float*    bnsq  = ws + o; o += 128;

    const int Etot       = GE + GN;
    const int edgeBlks   = (Etot + 255) / 256;
    const int edgeWvBlks = (Etot + 7) / 8;      // one wave per edge
    const int nodeWvBlks = (GN + 7) / 8;        // one wave per node

    // ---------------- Layer 1 ----------------
    wmma_gemm_kernel<<<GN / 16, 256, 0, stream>>>(x, W1, h1, F1, F1 / 16);
    attn_dots_kernel<<<nodeWvBlks, 256, 0, stream>>>(h1, att_src1, att_dst1,
                                                     a_src, a_dst, GN, F1, C1);
    hipMemsetAsync(amax,  0, (size_t)GN * 2 * 4, stream);
    hipMemsetAsync(denom, 0, (size_t)GN * 2 * 4, stream);
    hipMemsetAsync(out1,  0, (size_t)GN * F1 * 4, stream);
    edge_amax_kernel<<<edgeBlks, 256, 0, stream>>>(ei, a_src, a_dst, amax, GE, GN);
    edge_denom_kernel<<<edgeBlks, 256, 0, stream>>>(ei, a_src, a_dst, amax, denom, GE, GN);
    edge_aggregate_kernel<<<edgeWvBlks, 256, 0, stream>>>(
        ei, a_src, a_dst, amax, denom, h1, out1, GE, GN, F1, C1);

    // ---------------- BatchNorm ----------------
    hipMemsetAsync(bnsum, 0, 2 * 128 * 4, stream);   // bnsum + bnsq contiguous
    bn_stats_kernel<<<512, 128, 0, stream>>>(out1, bnsum, bnsq, GN);
    bn_apply_kernel<<<((size_t)GN * 128 + 255) / 256, 256, 0, stream>>>(
        out1, bnsum, bnsq, gamma, beta, GN);

    // ---------------- Layer 2 ----------------
    wmma_gemm_kernel<<<GN / 16, 256, 0, stream>>>(out1, W2, h2, F2, F2 / 16);
    attn_dots_kernel<<<nodeWvBlks, 256, 0, stream>>>(h2, att_src2, att_dst2,
                                                     a_src, a_dst, GN, F2, C2);
    hipMemsetAsync(amax,  0, (size_t)GN * 2 * 4, stream);
    hipMemsetAsync(denom, 0, (size_t)GN * 2 * 4, stream);
    hipMemsetAsync(out2,  0, (size_t)GN * F2 * 4, stream);
    edge_amax_kernel<<<edgeBlks, 256, 0, stream>>>(ei, a_src, a_dst, amax, GE, GN);
    edge_denom_kernel<<<edgeBlks, 256, 0, stream>>>(ei, a_src, a_dst, amax, denom, GE, GN);
    edge_aggregate_kernel<<<edgeWvBlks, 256, 0, stream>>>(
        ei, a_src, a_dst, amax, denom, h2, out2, GE, GN, F2, C2);

    // ---------------- Finalize ----------------
    finalize_kernel<<<nodeWvBlks, 256, 0, stream>>>(out2, b2, out, GN);
}